// StackMemory_34608846471639
// MI455X (gfx1250) — compile-verified
//
#include <hip/hip_runtime.h>
#include <hip/hip_bf16.h>
#include <math.h>

// Problem constants (from reference): B=2, S=2048, H=2048, NH=16, HD=128, SLOTS=16
#define BB 2
#define SS 2048
#define HH 2048
#define NHH 16
#define HDD 128
#define SLT 16
#define NA 48            // 3*NH logit columns
#define MROWS (BB*SS)    // 4096

typedef __attribute__((ext_vector_type(2))) float v2f;
typedef __attribute__((ext_vector_type(8))) float v8f;

// ---------------------------------------------------------------------------
// Kernel 1: logits = (hidden @ W_action^T + b) / sqrt(HD), via fp32 WMMA.
// One wave computes one 16x16 tile. grid = (M/16, N/16) = (256, 3).
// A (16x4 fp32) lane layout: lanes 0-15 -> K={k,k+1}; lanes 16-31 -> K={k+2,k+3}
// B (4x16 fp32): B[k][n] = W_action[n][k]  (K dim is W_action's contiguous dim)
// C (16x16 fp32): lane L, vgpr v -> M = v + 8*(L>>4), N = L&15
// ---------------------------------------------------------------------------
__global__ __launch_bounds__(32)
void action_logits_wmma(const float* __restrict__ hidden,
                        const float* __restrict__ Wa,
                        const float* __restrict__ ba,
                        float* __restrict__ logits)
{
    const int tileM = blockIdx.x;        // 0..255
    const int tileN = blockIdx.y;        // 0..2
    const int lane  = threadIdx.x;       // 0..31
    const int lo    = lane & 15;
    const int hi    = lane >> 4;         // 0 or 1

    const int row = tileM * 16 + lo;     // A row handled by this lane
    const int col = tileN * 16 + lo;     // B col handled by this lane

    const float* __restrict__ arow = hidden + (size_t)row * HH;
    const float* __restrict__ brow = Wa     + (size_t)col * HH;

    v8f c = {};
    #pragma unroll 4
    for (int k = 0; k < HH; k += 4) {
        const int kk = k + 2 * hi;
        v2f a, b;
        a.x = arow[kk]; a.y = arow[kk + 1];
        b.x = brow[kk]; b.y = brow[kk + 1];
        // (neg_a, A, neg_b, B, c_mod, C, reuse_a, reuse_b)
        c = __builtin_amdgcn_wmma_f32_16x16x4_f32(false, a, false, b,
                                                  (short)0, c, false, false);
    }

    const float scale = 0.08838834764831845f;   // 1/sqrt(128)
    const float bias  = ba[col];
    #pragma unroll
    for (int v = 0; v < 8; ++v) {
        const int m = tileM * 16 + v + 8 * hi;
        logits[(size_t)m * NA + col] = (c[v] + bias) * scale;
    }
}

// ---------------------------------------------------------------------------
// Kernel 2: fused stack update + gated readout.
// One wave (32 lanes) per (b,s,head); lane l owns head-dims [4l, 4l+4) as
// float4 -> all stack traffic moves as global_load/store_b128. The gate-score
// reduction is a pure intra-wave butterfly (wave covers all 128 dims), so no
// LDS and no barriers are needed; waves run fully independently.
// Block = 128 threads = 4 independent waves/heads.
// ---------------------------------------------------------------------------
__global__ __launch_bounds__(128)
void stack_update(const float* __restrict__ hidden,
                  const float* __restrict__ stack,
                  const float* __restrict__ mask,
                  const float* __restrict__ Wg,
                  const float* __restrict__ bg,
                  const float* __restrict__ resw,
                  const float* __restrict__ logits,
                  float* __restrict__ out,
                  float* __restrict__ new_stack,
                  float* __restrict__ new_mask)
{
    const int t    = threadIdx.x;
    const int wave = t >> 5;
    const int lane = t & 31;
    const int head = blockIdx.x * 4 + wave;     // 0 .. B*S*NH-1
    const int n    = head & (NHH - 1);
    const int bs   = head >> 4;

    const size_t hbase = (size_t)bs * HH + (size_t)n * HDD + 4 * lane;
    const size_t sbase = (size_t)head * (SLT * HDD) + 4 * lane;
    const size_t mbase = (size_t)head * SLT;
    const size_t lbase = (size_t)bs * NA + (size_t)n * 3;

    const float4 kval = *(const float4*)(hidden + hbase);

    float4 st[SLT];                      // loaded stack; overwritten with new_stack
    #pragma unroll
    for (int k = 0; k < SLT; ++k)
        st[k] = *(const float4*)(stack + sbase + k * HDD);

    // 3-way action softmax (uniform across the wave)
    const float l0 = logits[lbase + 0];
    const float l1 = logits[lbase + 1];
    const float l2 = logits[lbase + 2];
    const float mx3 = fmaxf(l0, fmaxf(l1, l2));
    const float e0 = __expf(l0 - mx3);
    const float e1 = __expf(l1 - mx3);
    const float e2 = __expf(l2 - mx3);
    const float inv3 = 1.0f / (e0 + e1 + e2);
    const float a_push = e0 * inv3;
    const float a_pop  = e1 * inv3;
    const float a_stay = e2 * inv3;

    const float4 wg = *(const float4*)(Wg + 4 * lane);

    // new_stack[k] = a_push*prev + a_pop*next + a_stay*cur, in place, with
    // immediate b128 store and gate-score partial per slot.
    float gp[SLT];
    float4 prev = kval;
    #pragma unroll
    for (int k = 0; k < SLT; ++k) {
        const float4 cur = st[k];
        float4 nxt;
        if (k == SLT - 1) { nxt.x = 0.f; nxt.y = 0.f; nxt.z = 0.f; nxt.w = 0.f; }
        else              { nxt = st[k + 1]; }
        float4 v;
        v.x = a_push * prev.x + a_pop * nxt.x + a_stay * cur.x;
        v.y = a_push * prev.y + a_pop * nxt.y + a_stay * cur.y;
        v.z = a_push * prev.z + a_pop * nxt.z + a_stay * cur.z;
        v.w = a_push * prev.w + a_pop * nxt.w + a_stay * cur.w;
        st[k] = v;
        prev  = cur;
        *(float4*)(new_stack + sbase + k * HDD) = v;
        gp[k] = v.x * wg.x + v.y * wg.y + v.z * wg.z + v.w * wg.w;
    }

    // full gate-score per slot via wave32 butterfly (all lanes end with sum)
    #pragma unroll
    for (int k = 0; k < SLT; ++k) {
        float v = gp[k];
        v += __shfl_xor(v, 16, 32);
        v += __shfl_xor(v,  8, 32);
        v += __shfl_xor(v,  4, 32);
        v += __shfl_xor(v,  2, 32);
        v += __shfl_xor(v,  1, 32);
        gp[k] = v;
    }

    // mask shift/update: lanes 0..15 own one slot each; broadcast via shuffles
    float nm = 0.0f;
    if (lane < SLT) {
        const float mk = mask[mbase + lane];
        const float mp = (lane == 0)       ? 1.0f : mask[mbase + lane - 1];
        const float mn = (lane == SLT - 1) ? 0.0f : mask[mbase + lane + 1];
        nm = a_push * mp + a_pop * mn + a_stay * mk;
        new_mask[mbase + lane] = nm;
    }

    // masked softmax over slots (per-lane, all-register)
    const float bg0 = bg[0];
    float smax = -3.0e38f;
    #pragma unroll
    for (int k = 0; k < SLT; ++k) {
        const float nmk = __shfl(nm, k, 32);
        float s = gp[k] + bg0 + (1.0f - nmk) * -1.0e9f;
        gp[k] = s;
        smax = fmaxf(smax, s);
    }
    float ssum = 0.0f;
    #pragma unroll
    for (int k = 0; k < SLT; ++k) { gp[k] = __expf(gp[k] - smax); ssum += gp[k]; }
    const float sinv = 1.0f / ssum;

    float4 mo; mo.x = 0.f; mo.y = 0.f; mo.z = 0.f; mo.w = 0.f;
    #pragma unroll
    for (int k = 0; k < SLT; ++k) {
        const float w = gp[k] * sinv;
        mo.x += w * st[k].x;
        mo.y += w * st[k].y;
        mo.z += w * st[k].z;
        mo.w += w * st[k].w;
    }

    const float rw = resw[0];
    float4 o;
    o.x = mo.x * rw + kval.x;
    o.y = mo.y * rw + kval.y;
    o.z = mo.z * rw + kval.z;
    o.w = mo.w * rw + kval.w;
    *(float4*)(out + hbase) = o;
}

// ---------------------------------------------------------------------------
extern "C" void kernel_launch(void* const* d_in, const int* in_sizes, int n_in,
                              void* d_out, int out_size, void* d_ws, size_t ws_size,
                              hipStream_t stream) {
    const float* hidden = (const float*)d_in[0];  // (B,S,H)
    const float* stack  = (const float*)d_in[1];  // (B,S,NH,SLOTS,HD)
    const float* mask   = (const float*)d_in[2];  // (B,S,NH,SLOTS)
    const float* Wa     = (const float*)d_in[3];  // (3*NH, H)
    const float* ba     = (const float*)d_in[4];  // (3*NH,)
    const float* Wg     = (const float*)d_in[5];  // (1, HD)
    const float* bg     = (const float*)d_in[6];  // (1,)
    const float* resw   = (const float*)d_in[7];  // (1,)

    float* out = (float*)d_out;                                   // (B,S,H)
    float* new_stack = out + (size_t)BB * SS * HH;                // (B,S,NH,SLOTS,HD)
    float* new_mask  = new_stack + (size_t)BB * SS * NHH * SLT * HDD; // (B,S,NH,SLOTS)

    float* logits = (float*)d_ws;   // (B*S, 48) fp32 = 768 KB

    dim3 g1(MROWS / 16, NA / 16, 1);     // (256, 3)
    action_logits_wmma<<<g1, 32, 0, stream>>>(hidden, Wa, ba, logits);

    dim3 g2(BB * SS * NHH / 4, 1, 1);    // 16384 blocks x 4 waves (1 head/wave)
    stack_update<<<g2, 128, 0, stream>>>(hidden, stack, mask, Wg, bg, resw,
                                         logits, out, new_stack, new_mask);
}